// MVHG_55920474194052
// MI455X (gfx1250) — compile-verified
//
#include <hip/hip_runtime.h>
#include <math.h>
#include <stdint.h>

#define NS   4096   // N_SAMPLES
#define NC   8      // N_CLASSES
#define MP1  513    // M_PER_CLASS + 1
#define NTH  256    // threads per block (8 wave32)

// ---------------------------------------------------------------------------
// CDNA5 async global->LDS path (ASYNCcnt). Guarded so compile never fails;
// fallback is a plain LDS store. On gfx1250 this lowers to
// global_load_async_to_lds_b32 + s_wait_asynccnt.
// ---------------------------------------------------------------------------
#if defined(__HIP_DEVICE_COMPILE__) && defined(__gfx1250__) && \
    __has_builtin(__builtin_amdgcn_global_load_async_to_lds_b32) && \
    __has_builtin(__builtin_amdgcn_s_wait_asynccnt)
#define USE_ASYNC_LDS 1
#else
#define USE_ASYNC_LDS 0
#endif

#if USE_ASYNC_LDS
typedef __attribute__((address_space(1))) int g_int;   // global
typedef __attribute__((address_space(3))) int l_int;   // LDS
#endif

__device__ __forceinline__ void async_cp_f32(const float* gsrc, float* lds_dst) {
#if USE_ASYNC_LDS
    __builtin_amdgcn_global_load_async_to_lds_b32(
        (g_int*)(uintptr_t)gsrc,
        (l_int*)(int*)lds_dst,
        0, 0);
#else
    *lds_dst = *gsrc;
#endif
}

__device__ __forceinline__ void async_wait_all() {
#if USE_ASYNC_LDS
    __builtin_amdgcn_s_wait_asynccnt(0);
#endif
}

__device__ __forceinline__ float sigm(float x) {
    return 1.0f / (1.0f + expf(-x));
}

// ---------------------------------------------------------------------------
// One block per sample. 8 classes processed sequentially (n_out dependency).
// ---------------------------------------------------------------------------
__global__ __launch_bounds__(NTH) void mvhg_fwd_kernel(
    const int*   __restrict__ m_all,     // (8,) int32
    const float* __restrict__ n_arr,     // (4096,1)
    const float* __restrict__ log_w,     // (4096,8)
    const float* __restrict__ temp_p,    // (1,)
    const float* __restrict__ gumbel,    // (7,4096,1,513)
    float*       __restrict__ out)       // y(8*4096*513) | x(8*4096) | ymask(8*4096*513) | logpX(4096)
{
    const int s   = blockIdx.x;
    const int tid = threadIdx.x;

    __shared__ float T[MP1];            // lgamma(x+1), x = 0..512
    __shared__ float Lsh[MP1];          // logits row (class i<7) / y row (class 7)
    __shared__ float gbuf[2][MP1];      // double-buffered gumbel row
    __shared__ float msh[MP1];          // ymask row for last class
    __shared__ float redA[NTH], redB[NTH];
    __shared__ int   redI[NTH];
    __shared__ float lwsh[NC];
    __shared__ float lwrest[NC - 1];
    __shared__ float mrest_sh[NC];
    __shared__ float mi_sh[NC];

    const size_t Y_BASE = 0;
    const size_t X_BASE = (size_t)NC * NS * MP1;
    const size_t M_BASE = X_BASE + (size_t)NC * NS;
    const size_t P_BASE = M_BASE + (size_t)NC * NS * MP1;

    // Kick off async prefetch of gumbel row for class 0 immediately.
    {
        const float* g0 = gumbel + (size_t)s * MP1;   // (0*NS + s)*513
        for (int x = tid; x < MP1; x += NTH) async_cp_f32(g0 + x, &gbuf[0][x]);
    }

    // lgamma table (covers lgamma(x+1) and, since m_i==512, lgamma(m_i-x+1)=T[512-x])
    for (int x = tid; x < MP1; x += NTH) T[x] = lgammaf((float)x + 1.0f);

    if (tid < NC) lwsh[tid] = log_w[(size_t)s * NC + tid];
    __syncthreads();

    // Per-class constants: m_i, m_rest, log_w_rest (logsumexp) — 8 threads do this.
    if (tid < NC) {
        mi_sh[tid] = (float)m_all[tid];
        float mr = 0.0f;
        for (int j = tid + 1; j < NC; ++j) mr += (float)m_all[j];
        mrest_sh[tid] = mr;
        if (tid < NC - 1) {
            const float lmr = logf(mr);
            float mx = -3.402823466e+38f;
            for (int j = tid + 1; j < NC; ++j) {
                float a = lwsh[j] + logf((float)m_all[j]) - lmr;
                mx = fmaxf(mx, a);
            }
            float sm = 0.0f;
            for (int j = tid + 1; j < NC; ++j) {
                float a = lwsh[j] + logf((float)m_all[j]) - lmr;
                sm += expf(a - mx);
            }
            lwrest[tid] = mx + logf(sm);
        }
    }
    __syncthreads();

    const float nS   = n_arr[s];
    const float temp = temp_p[0];
    float n_out = 0.0f;
    float logpX = 0.0f;

    // ------------------------- classes 0..6 -------------------------
    for (int i = 0; i < NC - 1; ++i) {
        const int cur = i & 1;

        // Wait for this class's gumbel row; barrier also fences previous
        // iteration's LDS consumers before we overwrite buffers below.
        async_wait_all();
        __syncthreads();

        // Overlap: stream next class's gumbel row into the other buffer.
        if (i + 1 < NC - 1) {
            const float* gn = gumbel + ((size_t)(i + 1) * NS + s) * MP1;
            for (int x = tid; x < MP1; x += NTH)
                async_cp_f32(gn + x, &gbuf[(i + 1) & 1][x]);
        }

        const float n_new  = fmaxf(nS - n_out, 0.0f);
        const float lwi    = lwsh[i];
        const float lwr    = lwrest[i];
        const float mrest  = mrest_sh[i];
        const float mi     = mi_sh[i];
        const bool  mi512  = (mi == 512.0f);

        float zreg[3], lreg[3];
        float lmaxz = -3.402823466e+38f;
        float lmaxl = -3.402823466e+38f;
        int   lidx  = tid;
        int   cnt   = 0;

        for (int k = 0; k < 3; ++k) {
            const int x = tid + k * NTH;
            if (x >= MP1) break;
            const float xf = (float)x;
            const float d  = n_new - xf;                 // n_new - x
            const float nx = fmaxf(d, 0.0f);             // relu
            const float hs = sigm(d) * sigm(mrest - nx);
            const float lg_mi_x = mi512 ? T[512 - x] : lgammaf(mi - xf + 1.0f);
            const float lgmx = T[x] + lgammaf(nx + 1.0f) + lg_mi_x
                             + lgammaf(fmaxf(mrest - nx, 0.0f) + 1.0f);
            const float logits = xf * lwi + d * lwr - lgmx
                               + 100.0f * logf(hs + 1e-20f);
            const float z = (logits + gbuf[cur][x]) / temp;
            Lsh[x]  = logits;
            zreg[k] = z;
            lreg[k] = logits;
            if (z > lmaxz) { lmaxz = z; lidx = x; }      // lowest index wins ties
            lmaxl = fmaxf(lmaxl, logits);
            ++cnt;
        }

        // Reduction 1: max+argmax of z (ties -> lowest index), max of logits.
        redA[tid] = lmaxz; redI[tid] = lidx; redB[tid] = lmaxl;
        __syncthreads();
        for (int off = NTH / 2; off > 0; off >>= 1) {
            if (tid < off) {
                const float ov = redA[tid + off];
                const int   oi = redI[tid + off];
                if (ov > redA[tid] || (ov == redA[tid] && oi < redI[tid])) {
                    redA[tid] = ov; redI[tid] = oi;
                }
                redB[tid] = fmaxf(redB[tid], redB[tid + off]);
            }
            __syncthreads();
        }
        const float zmax = redA[0];
        const int   a    = redI[0];
        const float lmax = redB[0];
        __syncthreads();

        // Reduction 2: sum exp(z - zmax) and sum exp(logits - lmax).
        float pz = 0.0f, pl = 0.0f;
        for (int k = 0; k < cnt; ++k) {
            pz += expf(zreg[k] - zmax);
            pl += expf(lreg[k] - lmax);
        }
        redA[tid] = pz; redB[tid] = pl;
        __syncthreads();
        for (int off = NTH / 2; off > 0; off >>= 1) {
            if (tid < off) { redA[tid] += redA[tid + off]; redB[tid] += redB[tid + off]; }
            __syncthreads();
        }
        const float sz = redA[0];
        const float sl = redB[0];

        // Straight-through value at argmax: v = p + (1 - p) (forward value of y).
        const float za    = (Lsh[a] + gbuf[cur][a]) / temp;
        const float psoft = expf(za - zmax) / sz;
        const float v     = psoft + (1.0f - psoft);
        logpX += ((Lsh[a] - lmax) - logf(sl)) * v;       // log_softmax(logits)[a] * v

        // y is exactly one-hot-like (non-argmax entries are exactly 0);
        // ymask (suffix cumsum) is v for x<=a, 0 after.
        float* yout = out + Y_BASE + ((size_t)i * NS + s) * MP1;
        float* mout = out + M_BASE + ((size_t)i * NS + s) * MP1;
        for (int x = tid; x < MP1; x += NTH) {
            yout[x] = (x == a) ? v : 0.0f;
            mout[x] = (x <= a) ? v : 0.0f;
        }
        const float xi = (float)(a + 1) * v - 1.0f;
        if (tid == 0) out[X_BASE + (size_t)i * NS + s] = xi;
        n_out = n_out + xi;
    }

    // ------------------------- last class (i = 7) -------------------------
    {
        __syncthreads();   // previous iteration's LDS consumers done
        const float n_new = fmaxf(nS - n_out, 0.0f);

        for (int k = 0; k < 3; ++k) {
            const int x = tid + k * NTH;
            if (x >= MP1) break;
            const float xf = (float)x;
            const float d  = n_new - xf;
            const float nx = fmaxf(d, 0.0f);
            Lsh[x] = sigm(d) * sigm(0.0f - nx);          // y_7 = hside (m_rest_b = 0)
        }
        __syncthreads();

        // Suffix cumsum in exact sequential order (matches flip/cumsum/flip).
        if (tid == 0) {
            float run = 0.0f;
            for (int x = MP1 - 1; x >= 0; --x) { run += Lsh[x]; msh[x] = run; }
        }
        __syncthreads();

        float ps = 0.0f;
        float* yout = out + Y_BASE + ((size_t)(NC - 1) * NS + s) * MP1;
        float* mout = out + M_BASE + ((size_t)(NC - 1) * NS + s) * MP1;
        for (int x = tid; x < MP1; x += NTH) {
            yout[x] = Lsh[x];
            const float mv = msh[x];
            mout[x] = mv;
            ps += mv;
        }
        redA[tid] = ps;
        __syncthreads();
        for (int off = NTH / 2; off > 0; off >>= 1) {
            if (tid < off) redA[tid] += redA[tid + off];
            __syncthreads();
        }
        if (tid == 0) {
            out[X_BASE + (size_t)(NC - 1) * NS + s] = redA[0] - 1.0f;  // x_7
            out[P_BASE + s] = logpX;                                   // log_p_X
        }
    }
}

extern "C" void kernel_launch(void* const* d_in, const int* in_sizes, int n_in,
                              void* d_out, int out_size, void* d_ws, size_t ws_size,
                              hipStream_t stream) {
    (void)in_sizes; (void)n_in; (void)out_size; (void)d_ws; (void)ws_size;
    const int*   m_all  = (const int*)d_in[0];
    const float* n_arr  = (const float*)d_in[1];
    const float* log_w  = (const float*)d_in[2];
    const float* temp   = (const float*)d_in[3];
    const float* gumbel = (const float*)d_in[4];
    float*       out    = (float*)d_out;

    mvhg_fwd_kernel<<<NS, NTH, 0, stream>>>(m_all, n_arr, log_w, temp, gumbel, out);
}